// Generator_36086315221707
// MI455X (gfx1250) — compile-verified
//
#include <hip/hip_runtime.h>
#include <hip/hip_bf16.h>

typedef __attribute__((ext_vector_type(2))) float v2f;
typedef __attribute__((ext_vector_type(8))) float v8f;

// D = A(16x4) * B(4x16) + C, fp32 WMMA. Layouts per CDNA5 ISA 7.12.2:
//  A: lanes 0-15 -> M=lane, VGPR0=K0,VGPR1=K1 ; lanes 16-31 -> K2,K3
//  B: lanes 0-15 -> N=lane, VGPR0=B[k0][n],VGPR1=B[k0+1][n]; lanes 16-31 -> k0+2,k0+3
//  C/D: VGPR r -> M=r (lanes 0-15) / M=r+8 (lanes 16-31), N=lane&15
__device__ __forceinline__ v8f wmma4(v2f a, v2f b, v8f c) {
  return __builtin_amdgcn_wmma_f32_16x16x4_f32(false, a, false, b, (short)0, c, false, false);
}

// ---------------- K-pair interleave rows of a KxN matrix: AP[(k/2)*N*2 + n*2 + (k&1)] ----------------
__global__ void pair_rows(const float* __restrict__ A, float* __restrict__ AP, int N, int total) {
  const int idx = blockIdx.x * 256 + threadIdx.x;
  if (idx >= total) return;
  const int p  = idx & 1;
  const int t0 = idx >> 1;
  const int n  = t0 % N;
  const int k2 = t0 / N;
  AP[idx] = A[(2 * k2 + p) * N + n];
}

// ---------------- Layer 1: GEMM M=256 (batch) x N=8192 (co*16+kh*4+kw) x K=100 ----------------
// W1P is K-pair interleaved: [k/2][8192][2]
__global__ void gemm_layer1(const float* __restrict__ Z, const float* __restrict__ W1P,
                            float* __restrict__ H1) {
  const int lane = threadIdx.x & 31;
  const int wave = threadIdx.x >> 5;
  const int half = lane >> 4;
  const int l16  = lane & 15;
  const int m0 = blockIdx.y * 16;
  const int n0 = blockIdx.x * 256 + wave * 64;
  v8f acc0 = {}, acc1 = {}, acc2 = {}, acc3 = {};
  const float* arow = Z + (m0 + l16) * 100;
  for (int k0 = 0; k0 < 100; k0 += 4) {
    const int ka = k0 + 2 * half;
    v2f a; a.x = arow[ka]; a.y = arow[ka + 1];
    const float* bp = W1P + ((k0 >> 1) + half) * 16384 + (n0 + l16) * 2;
    __builtin_prefetch(bp + 32768, 0, 3);
    v2f b0 = *(const v2f*)(bp);
    v2f b1 = *(const v2f*)(bp + 32);
    v2f b2 = *(const v2f*)(bp + 64);
    v2f b3 = *(const v2f*)(bp + 96);
    acc0 = wmma4(a, b0, acc0);
    acc1 = wmma4(a, b1, acc1);
    acc2 = wmma4(a, b2, acc2);
    acc3 = wmma4(a, b3, acc3);
  }
  for (int r = 0; r < 8; ++r) {
    const int m = m0 + r + 8 * half;
    float* o = H1 + m * 8192 + n0 + l16;
    o[0] = acc0[r]; o[16] = acc1[r]; o[32] = acc2[r]; o[48] = acc3[r];
  }
}

// ---------------- Weight reorder: w(Ci,Co,4,4) -> wr[kh][kw][ci/2][co][2] (K-pair interleaved) ----
__global__ void reorder_w_pair(const float* __restrict__ W, float* __restrict__ WR,
                               int Ci, int Co, int total) {
  const int idx = blockIdx.x * 256 + threadIdx.x;
  if (idx >= total) return;
  const int p   = idx & 1;
  const int t0  = idx >> 1;
  const int co  = t0 % Co;
  const int t1  = t0 / Co;
  const int c2  = t1 % (Ci >> 1);
  const int t   = t1 / (Ci >> 1);       // kh*4+kw
  WR[idx] = W[((2 * c2 + p) * Co + co) * 16 + t];
}

// ---------------- Transposed conv (k=4, s=2, p=1) as 4-parity implicit GEMM ----------------
// M = 256*Hi*Wi per parity, N = Co, K = 4 taps * Ci. wave tile 16x64.
__global__ void convt_gemm(const float* __restrict__ X, const float* __restrict__ WRP,
                           float* __restrict__ Y, int Ci, int Hi, int Wi, int Co) {
  const int lane = threadIdx.x & 31;
  const int wave = threadIdx.x >> 5;
  const int half = lane >> 4;
  const int l16  = lane & 15;
  const int ph = blockIdx.z >> 1, pw = blockIdx.z & 1;
  const int n0 = blockIdx.x * 64;
  const int m0 = (blockIdx.y * 4 + wave) * 16;
  const int HW = Hi * Wi;

  const int m  = m0 + l16;
  const int b  = m / HW;
  const int yx = m - b * HW;
  const int yy = yx / Wi;
  const int xx = yx - yy * Wi;

  // oh = 2*ih - 1 + kh : ph=0 -> kh{1,3}, ih{yy,yy-1} ; ph=1 -> kh{0,2}, ih{yy+1,yy}
  int ih[2], kh[2], iw[2], kw[2];
  if (ph == 0) { kh[0] = 1; ih[0] = yy;     kh[1] = 3; ih[1] = yy - 1; }
  else         { kh[0] = 0; ih[0] = yy + 1; kh[1] = 2; ih[1] = yy;     }
  if (pw == 0) { kw[0] = 1; iw[0] = xx;     kw[1] = 3; iw[1] = xx - 1; }
  else         { kw[0] = 0; iw[0] = xx + 1; kw[1] = 2; iw[1] = xx;     }

  v8f acc0 = {}, acc1 = {}, acc2 = {}, acc3 = {};
#pragma unroll
  for (int t = 0; t < 4; ++t) {
    const int th = t >> 1, tw = t & 1;
    const int ihh = ih[th], iww = iw[tw];
    const bool valid = (ihh >= 0) && (ihh < Hi) && (iww >= 0) && (iww < Wi);
    const float* ap = X + ((b * Ci) * Hi + ihh) * Wi + iww;        // + ci*HW
    const float* bbase = WRP + ((kh[th] * 4 + kw[tw]) * (Ci >> 1) * Co + n0 + l16) * 2;
    for (int c0 = 0; c0 < Ci; c0 += 4) {
      const int ca = c0 + 2 * half;
      v2f a; a.x = 0.f; a.y = 0.f;
      if (valid) { a.x = ap[ca * HW]; a.y = ap[(ca + 1) * HW]; }
      const float* bp = bbase + ((c0 >> 1) + half) * (Co * 2);
      __builtin_prefetch(bp + 4 * Co, 0, 3);
      v2f b0 = *(const v2f*)(bp);
      v2f b1 = *(const v2f*)(bp + 32);
      v2f b2 = *(const v2f*)(bp + 64);
      v2f b3 = *(const v2f*)(bp + 96);
      acc0 = wmma4(a, b0, acc0);
      acc1 = wmma4(a, b1, acc1);
      acc2 = wmma4(a, b2, acc2);
      acc3 = wmma4(a, b3, acc3);
    }
  }
  const int Ho = 2 * Hi, Wo = 2 * Wi, HoWo = Ho * Wo;
  for (int r = 0; r < 8; ++r) {
    const int mm = m0 + r + 8 * half;
    const int bb = mm / HW;
    const int yx2 = mm - bb * HW;
    const int y2 = yx2 / Wi;
    const int x2 = yx2 - y2 * Wi;
    const int addr = (bb * Co + n0 + l16) * HoWo + (2 * y2 + ph) * Wo + (2 * x2 + pw);
    Y[addr]             = acc0[r];
    Y[addr + 16 * HoWo] = acc1[r];
    Y[addr + 32 * HoWo] = acc2[r];
    Y[addr + 48 * HoWo] = acc3[r];
  }
}

// ---------------- BatchNorm (training stats) — deterministic 2-stage ----------------
__global__ void bn_stats_partial(const float* __restrict__ X, float* __restrict__ partial,
                                 int C, int HW, int bPerSlice, int nSlices) {
  const int c = blockIdx.x;
  const int sl = blockIdx.y;
  const int bStart = sl * bPerSlice;
  const int total = bPerSlice * HW;
  float s = 0.f, s2 = 0.f;
  for (int idx = threadIdx.x; idx < total; idx += 256) {
    const int bb = bStart + idx / HW;
    const int i = idx - (idx / HW) * HW;
    const float v = X[(bb * C + c) * HW + i];
    s += v; s2 += v * v;
  }
  __shared__ float r1[256], r2[256];
  r1[threadIdx.x] = s; r2[threadIdx.x] = s2;
  __syncthreads();
  for (int st = 128; st > 0; st >>= 1) {
    if (threadIdx.x < st) {
      r1[threadIdx.x] += r1[threadIdx.x + st];
      r2[threadIdx.x] += r2[threadIdx.x + st];
    }
    __syncthreads();
  }
  if (threadIdx.x == 0) {
    partial[(c * nSlices + sl) * 2]     = r1[0];
    partial[(c * nSlices + sl) * 2 + 1] = r2[0];
  }
}

__global__ void bn_finalize(const float* __restrict__ partial, const float* __restrict__ gamma,
                            const float* __restrict__ beta, float* __restrict__ scsh,
                            int C, int nSlices, float invN) {
  const int c = blockIdx.x * 256 + threadIdx.x;
  if (c >= C) return;
  float s = 0.f, s2 = 0.f;
  for (int i = 0; i < nSlices; ++i) {
    s  += partial[(c * nSlices + i) * 2];
    s2 += partial[(c * nSlices + i) * 2 + 1];
  }
  const float m = s * invN;
  const float v = s2 * invN - m * m;
  const float sc = gamma[c] * rsqrtf(v + 1e-5f);
  scsh[2 * c]     = sc;
  scsh[2 * c + 1] = beta[c] - m * sc;
}

__global__ void bn_relu_apply(float* __restrict__ X, const float* __restrict__ scsh,
                              int C, int HW, int total) {
  const int i = blockIdx.x * 256 + threadIdx.x;
  if (i >= total) return;
  const int c = (i / HW) % C;
  const float v = X[i] * scsh[2 * c] + scsh[2 * c + 1];
  X[i] = fmaxf(v, 0.f);
}

// ---------------- Haar DWT features -> K-pair interleaved transposed matrix ----------------
// FTP[f/2][b][2] holds feats[b][f]; grid (64 i, 4 c4, 4 b-tiles of 64), block 256.
__global__ void haar_kernel(const float* __restrict__ X, float* __restrict__ FTP) {
  __shared__ float lds[4][64][65];
  const int i  = blockIdx.x;        // output row 0..63
  const int c4 = blockIdx.y;        // 0..3
  const int b0 = blockIdx.z * 64;   // batch tile
  const int j  = threadIdx.x & 63;
  const int br = threadIdx.x >> 6;  // 0..3
  for (int blc = br; blc < 64; blc += 4) {
    const int bb = b0 + blc;
    const float* p = X + bb * 65536 + c4 * 16384 + (2 * i) * 128 + 2 * j;
    const float a = p[0], b = p[1], c = p[128], d = p[129];
    lds[0][j][blc] = (a + b + c + d) * 0.5f;
    lds[1][j][blc] = (a + b - c - d) * 0.5f;
    lds[2][j][blc] = (a - b + c - d) * 0.5f;
    lds[3][j][blc] = (a - b - c + d) * 0.5f;
  }
  __syncthreads();
  const int blw = threadIdx.x & 63;
  const int s   = threadIdx.x >> 6;  // subband
  const int fbase = ((c4 * 4 + s) * 64 + i) * 64;   // even
  for (int j2 = 0; j2 < 64; j2 += 2) {
    v2f v; v.x = lds[s][j2][blw]; v.y = lds[s][j2 + 1][blw];
    *(v2f*)(FTP + ((fbase + j2) >> 1) * 512 + (b0 + blw) * 2) = v;
  }
}

// ---------------- Linear: D[j(128), b(256)] = sum_k WL[j,k]*FT[k,b], split-K (32 chunks of 2048) ----
__global__ void linear_stage1(const float* __restrict__ WL, const float* __restrict__ FTP,
                              float* __restrict__ PART) {
  const int lane = threadIdx.x & 31;
  const int wave = threadIdx.x >> 5;
  const int half = lane >> 4;
  const int l16  = lane & 15;
  const int n0 = blockIdx.x * 64;                  // batch tile
  const int m0 = (blockIdx.y * 4 + wave) * 16;     // output-feature tile
  const int kStart = blockIdx.z * 2048;
  v8f acc0 = {}, acc1 = {}, acc2 = {}, acc3 = {};
  const float* arow = WL + (m0 + l16) * 65536;
  for (int k0 = kStart; k0 < kStart + 2048; k0 += 4) {
    const int ka = k0 + 2 * half;
    v2f a; a.x = arow[ka]; a.y = arow[ka + 1];
    const float* bp = FTP + ((k0 >> 1) + half) * 512 + (n0 + l16) * 2;
    __builtin_prefetch(bp + 1024, 0, 3);
    v2f b0 = *(const v2f*)(bp);
    v2f b1 = *(const v2f*)(bp + 32);
    v2f b2 = *(const v2f*)(bp + 64);
    v2f b3 = *(const v2f*)(bp + 96);
    acc0 = wmma4(a, b0, acc0);
    acc1 = wmma4(a, b1, acc1);
    acc2 = wmma4(a, b2, acc2);
    acc3 = wmma4(a, b3, acc3);
  }
  float* o = PART + blockIdx.z * 32768;
  for (int r = 0; r < 8; ++r) {
    const int m = m0 + r + 8 * half;
    float* p = o + m * 256 + n0 + l16;
    p[0] = acc0[r]; p[16] = acc1[r]; p[32] = acc2[r]; p[48] = acc3[r];
  }
}

__global__ void linear_stage2(const float* __restrict__ PART, const float* __restrict__ BL,
                              float* __restrict__ OUT) {
  const int idx = blockIdx.x * 256 + threadIdx.x;   // j*256 + b
  if (idx >= 32768) return;
  const int j = idx >> 8;
  const int b = idx & 255;
  float s = 0.f;
  for (int z = 0; z < 32; ++z) s += PART[z * 32768 + idx];
  OUT[b * 128 + j] = tanhf(s + BL[j]);
}

extern "C" void kernel_launch(void* const* d_in, const int* in_sizes, int n_in,
                              void* d_out, int out_size, void* d_ws, size_t ws_size,
                              hipStream_t stream) {
  const float* z  = (const float*)d_in[0];
  const float* w1 = (const float*)d_in[1];
  const float* w2 = (const float*)d_in[2];
  const float* w3 = (const float*)d_in[3];
  const float* w4 = (const float*)d_in[4];
  const float* g1 = (const float*)d_in[5];
  const float* b1 = (const float*)d_in[6];
  const float* g2 = (const float*)d_in[7];
  const float* b2 = (const float*)d_in[8];
  const float* g3 = (const float*)d_in[9];
  const float* b3 = (const float*)d_in[10];
  const float* g4 = (const float*)d_in[11];
  const float* b4 = (const float*)d_in[12];
  const float* wl = (const float*)d_in[13];
  const float* blin = (const float*)d_in[14];
  float* out = (float*)d_out;

  char* ws = (char*)d_ws;
  float* partBN = (float*)(ws);                          //  64 KB (512ch x 16sl x 2)
  float* scsh   = (float*)(ws + 65536);                  //   4 KB
  float* partK  = (float*)(ws + 131072);                 //   4 MB split-K partials
  float* bufW   = (float*)(ws + 131072 + 4194304);       //   8 MB weight reorder
  float* bufA   = (float*)(ws + 131072 + 4194304 + 8388608); // 64 MB
  float* bufB   = bufA + 16777216;                       //  64 MB

  // Layer 1: (256,100,1,1) -> (256,512,4,4)
  pair_rows<<<dim3(819200 / 256, 1, 1), 256, 0, stream>>>(w1, bufW, 8192, 819200);
  gemm_layer1<<<dim3(32, 16, 1), 128, 0, stream>>>(z, bufW, bufA);
  bn_stats_partial<<<dim3(512, 8, 1), 256, 0, stream>>>(bufA, partBN, 512, 16, 32, 8);
  bn_finalize<<<dim3(2, 1, 1), 256, 0, stream>>>(partBN, g1, b1, scsh, 512, 8, 1.f / 4096.f);
  bn_relu_apply<<<dim3(2097152 / 256, 1, 1), 256, 0, stream>>>(bufA, scsh, 512, 16, 2097152);

  // Layer 2: -> (256,256,8,8)
  reorder_w_pair<<<dim3(2097152 / 256, 1, 1), 256, 0, stream>>>(w2, bufW, 512, 256, 2097152);
  convt_gemm<<<dim3(4, 64, 4), 128, 0, stream>>>(bufA, bufW, bufB, 512, 4, 4, 256);
  bn_stats_partial<<<dim3(256, 8, 1), 256, 0, stream>>>(bufB, partBN, 256, 64, 32, 8);
  bn_finalize<<<dim3(1, 1, 1), 256, 0, stream>>>(partBN, g2, b2, scsh, 256, 8, 1.f / 16384.f);
  bn_relu_apply<<<dim3(4194304 / 256, 1, 1), 256, 0, stream>>>(bufB, scsh, 256, 64, 4194304);

  // Layer 3: -> (256,128,16,16)
  reorder_w_pair<<<dim3(524288 / 256, 1, 1), 256, 0, stream>>>(w3, bufW, 256, 128, 524288);
  convt_gemm<<<dim3(2, 256, 4), 128, 0, stream>>>(bufB, bufW, bufA, 256, 8, 8, 128);
  bn_stats_partial<<<dim3(128, 16, 1), 256, 0, stream>>>(bufA, partBN, 128, 256, 16, 16);
  bn_finalize<<<dim3(1, 1, 1), 256, 0, stream>>>(partBN, g3, b3, scsh, 128, 16, 1.f / 65536.f);
  bn_relu_apply<<<dim3(8388608 / 256, 1, 1), 256, 0, stream>>>(bufA, scsh, 128, 256, 8388608);

  // Layer 4: -> (256,64,32,32)
  reorder_w_pair<<<dim3(131072 / 256, 1, 1), 256, 0, stream>>>(w4, bufW, 128, 64, 131072);
  convt_gemm<<<dim3(1, 1024, 4), 128, 0, stream>>>(bufA, bufW, bufB, 128, 16, 16, 64);
  bn_stats_partial<<<dim3(64, 16, 1), 256, 0, stream>>>(bufB, partBN, 64, 1024, 16, 16);
  bn_finalize<<<dim3(1, 1, 1), 256, 0, stream>>>(partBN, g4, b4, scsh, 64, 16, 1.f / 262144.f);
  bn_relu_apply<<<dim3(16777216 / 256, 1, 1), 256, 0, stream>>>(bufB, scsh, 64, 1024, 16777216);

  // Haar DWT -> FTP (K-pair interleaved transposed features) in bufA
  haar_kernel<<<dim3(64, 4, 4), 256, 0, stream>>>(bufB, bufA);

  // Linear + tanh
  linear_stage1<<<dim3(4, 2, 32), 128, 0, stream>>>(wl, bufA, partK);
  linear_stage2<<<dim3(128, 1, 1), 256, 0, stream>>>(partK, blin, out);
}